// CombinedCriterionAEImpulse_30923764531286
// MI455X (gfx1250) — compile-verified
//
#include <hip/hip_runtime.h>
#include <hip/hip_bf16.h>
#include <math.h>

// Problem sizes (fixed by the reference setup_inputs()).
#define NPRED 8192
#define NGT   32768
#define FDIM  6    // row stride of pred_feat / gt_data

typedef float v2f __attribute__((ext_vector_type(2)));
typedef float v8f __attribute__((ext_vector_type(8)));

// ---------------------------------------------------------------------------
// One 16x16 tile of pairwise squared distances via V_WMMA_F32_16X16X4_F32,
// folded into a running per-row min/argmin.
// A operand: pred tile 16x4 (x,y,z,0); B operand: 16 points as 4x16 columns.
// C/D layout (ISA 7.12.2): VGPR r holds M=r (lanes 0-15) / M=r+8 (lanes
// 16-31); N = lane & 15.
// ---------------------------------------------------------------------------
__device__ __forceinline__ void tile_min_argmin(float gx, float gy, float gz,
                                                int col, int half, v2f a,
                                                const float px2[8],
                                                float bd2[8], int bidx[8]) {
  const float q = gx * gx + gy * gy + gz * gz;   // |g|^2 for column N
  v2f b;
  b.x = (half == 0) ? gx : gz;
  b.y = (half == 0) ? gy : 0.0f;
  v8f c = {0.f, 0.f, 0.f, 0.f, 0.f, 0.f, 0.f, 0.f};
  c = __builtin_amdgcn_wmma_f32_16x16x4_f32(
      /*neg_a=*/false, a, /*neg_b=*/false, b,
      /*c_mod=*/(short)0, c, /*reuse_a=*/false, /*reuse_b=*/false);
#pragma unroll
  for (int r = 0; r < 8; ++r) {
    float d2 = fmaxf(fmaf(-2.0f, c[r], px2[r] + q), 0.0f);
    bool better = d2 < bd2[r];
    bd2[r]  = better ? d2  : bd2[r];
    bidx[r] = better ? col : bidx[r];
  }
}

// Same, but min-only with self-column exclusion (repulsion term).
__device__ __forceinline__ void tile_min_noself(float gx, float gy, float gz,
                                                int col, int half, int m0, v2f a,
                                                const float px2[8],
                                                float bd2[8]) {
  const float q = gx * gx + gy * gy + gz * gz;
  v2f b;
  b.x = (half == 0) ? gx : gz;
  b.y = (half == 0) ? gy : 0.0f;
  v8f c = {0.f, 0.f, 0.f, 0.f, 0.f, 0.f, 0.f, 0.f};
  c = __builtin_amdgcn_wmma_f32_16x16x4_f32(
      false, a, false, b, (short)0, c, false, false);
#pragma unroll
  for (int r = 0; r < 8; ++r) {
    const int row = m0 + half * 8 + r;
    float d2 = fmaxf(fmaf(-2.0f, c[r], px2[r] + q), 0.0f);
    d2 = (col == row) ? 3.0e38f : d2;  // exclude diagonal (the 1e6*eye term)
    bd2[r] = fminf(bd2[r], d2);
  }
}

// ---------------------------------------------------------------------------
// Kernel A: nearest-neighbor pred -> gt. One wave per block owns a 16-row
// pred tile and a CHUNK-column gt slice. B tiles are register double-buffered
// two-at-a-time so the global_load_b96 latency overlaps the WMMA + min
// epilogue of the previous pair instead of stalling each tile.
// ---------------------------------------------------------------------------
template <int CHUNK>
__global__ __launch_bounds__(32)
void nn_wmma_kernel(const float* __restrict__ pred,
                    const float* __restrict__ gt,
                    unsigned long long* __restrict__ best) {
  const int m0   = blockIdx.x * 16;
  const int lane = threadIdx.x;
  const int half = lane >> 4;   // 0: K0/K1 + M0..7 results; 1: K2/K3 + M8..15
  const int l    = lane & 15;

  const float* pr = pred + (size_t)(m0 + l) * FDIM;
  v2f a;
  a.x = (half == 0) ? pr[0] : pr[2];
  a.y = (half == 0) ? pr[1] : 0.0f;

  float px2[8];
#pragma unroll
  for (int r = 0; r < 8; ++r) {
    const float* p = pred + (size_t)(m0 + half * 8 + r) * FDIM;
    px2[r] = p[0] * p[0] + p[1] * p[1] + p[2] * p[2];
  }

  float bd2[8];
  int   bidx[8];
#pragma unroll
  for (int r = 0; r < 8; ++r) { bd2[r] = 3.0e38f; bidx[r] = 0; }

  const int nbeg = blockIdx.y * CHUNK;

  // Prime the double buffer with the first two 16-column tiles.
  const float* g0 = gt + (size_t)(nbeg + l) * FDIM;
  const float* g1 = g0 + 16 * FDIM;
  float cx0 = g0[0], cy0 = g0[1], cz0 = g0[2];
  float cx1 = g1[0], cy1 = g1[1], cz1 = g1[2];

  for (int n0 = nbeg; n0 < nbeg + CHUNK; n0 += 32) {
    // Issue next pair of loads before consuming the current pair.
    const int nn  = n0 + 32;
    const int pre = (nn < nbeg + CHUNK) ? nn : nbeg;   // harmless tail reload
    const float* h0 = gt + (size_t)(pre + l) * FDIM;
    const float* h1 = h0 + 16 * FDIM;
    const float nx0 = h0[0], ny0 = h0[1], nz0 = h0[2];
    const float nx1 = h1[0], ny1 = h1[1], nz1 = h1[2];

    tile_min_argmin(cx0, cy0, cz0, n0 + l,      half, a, px2, bd2, bidx);
    tile_min_argmin(cx1, cy1, cz1, n0 + 16 + l, half, a, px2, bd2, bidx);

    cx0 = nx0; cy0 = ny0; cz0 = nz0;
    cx1 = nx1; cy1 = ny1; cz1 = nz1;
  }

  // Min/argmin across the 16 lanes of each half (xor masks < 16 stay in-half).
#pragma unroll
  for (int m = 8; m >= 1; m >>= 1) {
#pragma unroll
    for (int r = 0; r < 8; ++r) {
      float od = __shfl_xor(bd2[r], m, 32);
      int   oi = __shfl_xor(bidx[r], m, 32);
      bool take = (od < bd2[r]) || (od == bd2[r] && oi < bidx[r]);
      bd2[r]  = take ? od : bd2[r];
      bidx[r] = take ? oi : bidx[r];
    }
  }

  if (l == 0) {
#pragma unroll
    for (int r = 0; r < 8; ++r) {
      const int row = m0 + half * 8 + r;
      // Pack (d2 bits | index): integer atomicMin => deterministic, ties
      // resolve to the smallest index (matches jnp.argmin first-hit rule).
      unsigned long long pk =
          ((unsigned long long)__float_as_uint(bd2[r]) << 32) |
          (unsigned long long)(unsigned)bidx[r];
      atomicMin(&best[row], pk);
    }
  }
}

// ---------------------------------------------------------------------------
// Kernel B: repulsion min distance pred -> pred, same pipelined WMMA tiling.
// ---------------------------------------------------------------------------
template <int CHUNK>
__global__ __launch_bounds__(32)
void rep_wmma_kernel(const float* __restrict__ pred,
                     unsigned* __restrict__ repmin) {
  const int m0   = blockIdx.x * 16;
  const int lane = threadIdx.x;
  const int half = lane >> 4;
  const int l    = lane & 15;

  const float* pr = pred + (size_t)(m0 + l) * FDIM;
  v2f a;
  a.x = (half == 0) ? pr[0] : pr[2];
  a.y = (half == 0) ? pr[1] : 0.0f;

  float px2[8];
#pragma unroll
  for (int r = 0; r < 8; ++r) {
    const float* p = pred + (size_t)(m0 + half * 8 + r) * FDIM;
    px2[r] = p[0] * p[0] + p[1] * p[1] + p[2] * p[2];
  }

  float bd2[8];
#pragma unroll
  for (int r = 0; r < 8; ++r) bd2[r] = 3.0e38f;

  const int nbeg = blockIdx.y * CHUNK;

  const float* g0 = pred + (size_t)(nbeg + l) * FDIM;
  const float* g1 = g0 + 16 * FDIM;
  float cx0 = g0[0], cy0 = g0[1], cz0 = g0[2];
  float cx1 = g1[0], cy1 = g1[1], cz1 = g1[2];

  for (int n0 = nbeg; n0 < nbeg + CHUNK; n0 += 32) {
    const int nn  = n0 + 32;
    const int pre = (nn < nbeg + CHUNK) ? nn : nbeg;
    const float* h0 = pred + (size_t)(pre + l) * FDIM;
    const float* h1 = h0 + 16 * FDIM;
    const float nx0 = h0[0], ny0 = h0[1], nz0 = h0[2];
    const float nx1 = h1[0], ny1 = h1[1], nz1 = h1[2];

    tile_min_noself(cx0, cy0, cz0, n0 + l,      half, m0, a, px2, bd2);
    tile_min_noself(cx1, cy1, cz1, n0 + 16 + l, half, m0, a, px2, bd2);

    cx0 = nx0; cy0 = ny0; cz0 = nz0;
    cx1 = nx1; cy1 = ny1; cz1 = nz1;
  }

#pragma unroll
  for (int m = 8; m >= 1; m >>= 1) {
#pragma unroll
    for (int r = 0; r < 8; ++r)
      bd2[r] = fminf(bd2[r], __shfl_xor(bd2[r], m, 32));
  }

  if (l == 0) {
#pragma unroll
    for (int r = 0; r < 8; ++r) {
      const int row = m0 + half * 8 + r;
      // bd2 >= 0 => float bits order like unsigned ints.
      atomicMin(&repmin[row], __float_as_uint(bd2[r]));
    }
  }
}

// ---------------------------------------------------------------------------
// Workspace init (ws/out are poisoned by the harness; must re-init each call).
// ---------------------------------------------------------------------------
__global__ void init_kernel(unsigned long long* __restrict__ best,
                            unsigned* __restrict__ repmin,
                            float* __restrict__ partials,
                            float* __restrict__ out) {
  const int i = blockIdx.x * blockDim.x + threadIdx.x;
  if (i < NPRED) {
    best[i]   = ~0ull;
    repmin[i] = 0xFFFFFFFFu;
  }
  if (i < 32) partials[i] = 0.0f;
  if (i == 0) out[0] = 0.0f;
}

// ---------------------------------------------------------------------------
// Kernel C: per-point loss terms + fixed-order block reduction to partials.
// ---------------------------------------------------------------------------
__global__ __launch_bounds__(256)
void finalize_kernel(const float* __restrict__ pred,
                     const float* __restrict__ gt,
                     const unsigned long long* __restrict__ best,
                     const unsigned* __restrict__ repmin,
                     float* __restrict__ partials) {
  __shared__ float sm[256];
  const int i = blockIdx.x * 256 + threadIdx.x;

  const float* p = pred + (size_t)i * FDIM;
  const unsigned long long pk = best[i];
  const int idx = (int)(unsigned)(pk & 0xFFFFFFFFull);
  const float* g = gt + (size_t)idx * FDIM;

  // Attraction: mean over (NPRED, 3) of squared diffs.
  const float dx = p[0] - g[0], dy = p[1] - g[1], dz = p[2] - g[2];
  const float attract = dx * dx + dy * dy + dz * dz;

  // Normal consistency.
  const float pnx = p[3], pny = p[4], pnz = p[5];
  const float gnx = g[3], gny = g[4], gnz = g[5];
  const float pl = fmaxf(sqrtf(pnx * pnx + pny * pny + pnz * pnz), 1e-5f);
  const float gl = fmaxf(sqrtf(gnx * gnx + gny * gny + gnz * gnz), 1e-5f);
  const float cosv = (pnx * gnx + pny * gny + pnz * gnz) / (pl * gl);

  // Repulsion: softplus(100*(0.3 - min_dist))^2, numerically stable softplus.
  const float md = sqrtf(__uint_as_float(repmin[i]));
  const float x = 100.0f * (0.3f - md);
  const float sp = (x > 20.0f) ? x : log1pf(expf(x));

  const float val = attract * (1.0f / (NPRED * 3.0f))
                  + sp * sp * (1.0f / NPRED)
                  + 10.0f * (1.0f - cosv) * (1.0f / NPRED);

  sm[threadIdx.x] = val;
  __syncthreads();
#pragma unroll
  for (int s = 128; s >= 1; s >>= 1) {
    if (threadIdx.x < s) sm[threadIdx.x] += sm[threadIdx.x + s];
    __syncthreads();
  }
  if (threadIdx.x == 0) partials[blockIdx.x] = sm[0];
}

// Kernel D: fixed shuffle-tree sum of the 32 block partials (deterministic).
__global__ __launch_bounds__(32)
void sum_kernel(const float* __restrict__ partials, float* __restrict__ out) {
  float v = partials[threadIdx.x];
#pragma unroll
  for (int m = 16; m >= 1; m >>= 1) v += __shfl_xor(v, m, 32);
  if (threadIdx.x == 0) out[0] = v;
}

// ---------------------------------------------------------------------------
extern "C" void kernel_launch(void* const* d_in, const int* in_sizes, int n_in,
                              void* d_out, int out_size, void* d_ws, size_t ws_size,
                              hipStream_t stream) {
  const float* pred = (const float*)d_in[0];   // pred_feat (8192,6)
  // d_in[1] (pred_decoder) and d_in[2] (input_data) are unused by the reference.
  const float* gt   = (const float*)d_in[3];   // gt_data  (32768,6)
  float* out = (float*)d_out;

  char* ws = (char*)d_ws;
  unsigned long long* best = (unsigned long long*)ws;                 // 8192 * 8B
  unsigned* repmin         = (unsigned*)(ws + (size_t)NPRED * 8);     // 8192 * 4B
  float* partials          = (float*)(ws + (size_t)NPRED * 8 + (size_t)NPRED * 4);

  init_kernel<<<(NPRED + 255) / 256, 256, 0, stream>>>(best, repmin, partials, out);

  // NN search: 512 pred tiles x 16 gt chunks (chunk = 2048 cols = 128 WMMA tiles).
  dim3 gA(NPRED / 16, NGT / 2048);
  nn_wmma_kernel<2048><<<gA, 32, 0, stream>>>(pred, gt, best);

  // Repulsion: 512 pred tiles x 8 pred chunks (chunk = 1024 cols = 64 WMMA tiles).
  dim3 gB(NPRED / 16, NPRED / 1024);
  rep_wmma_kernel<1024><<<gB, 32, 0, stream>>>(pred, repmin);

  finalize_kernel<<<NPRED / 256, 256, 0, stream>>>(pred, gt, best, repmin, partials);
  sum_kernel<<<1, 32, 0, stream>>>(partials, out);
}